// GCN_45028437131774
// MI455X (gfx1250) — compile-verified
//
#include <hip/hip_runtime.h>
#include <cstdint>

#define NUM_USER 50000
#define NUM_ITEM 50000
#define NTOT     (NUM_USER + NUM_ITEM)   // 100000, multiple of 16
#define DIM_FEAT 128
#define DIM      64
#define NEDGE    2000000
#define NEG_SLOPE 0.01f

typedef __attribute__((ext_vector_type(2))) float v2f;
typedef __attribute__((ext_vector_type(8))) float v8f;

__device__ __forceinline__ float leaky(float x) { return x > 0.0f ? x : x * NEG_SLOPE; }

// ---------------------------------------------------------------------------
// Generic fp32 WMMA GEMM: C[rows x 64] = act( A[rows x KA] @ B + bias )
//   TRANSB=false : B = W stored [KA x 64] row-major      (conv_w, [in,out])
//   TRANSB=true  : B = W^T, W stored [64 x KA] row-major (nn.Linear [out,in])
// One wave per 16-row tile, 4 N-tiles of 16, K stepped by 4 via
// V_WMMA_F32_16X16X4_F32.  A-frag: lane L holds row (L&15), k = kk+(L>>4)*2+{0,1}.
// C-frag: VGPR v -> row (L>>4)*8+v, col = ntile*16 + (L&15).
// ---------------------------------------------------------------------------
template<int KA, bool TRANSB, bool LEAKY, bool BIAS>
__global__ __launch_bounds__(256) void gemm_wmma(
    const float* __restrict__ A, const float* __restrict__ W,
    const float* __restrict__ bias, float* __restrict__ C, int ntiles)
{
  const int wave = threadIdx.x >> 5;
  const int lane = threadIdx.x & 31;
  int tile = blockIdx.x * 8 + wave;
  if (tile >= ntiles) tile = ntiles - 1;     // wave-uniform clamp: EXEC stays all-1 for WMMA
  const int r  = lane & 15;
  const int kh = lane >> 4;

  const float* Arow = A + (size_t)(tile * 16 + r) * KA;

  v8f acc[4];
#pragma unroll
  for (int nt = 0; nt < 4; ++nt)
#pragma unroll
    for (int j = 0; j < 8; ++j) acc[nt][j] = 0.0f;

#pragma unroll
  for (int kk = 0; kk < KA; kk += 4) {
    const int ka = kk + kh * 2;
    v2f a;
    a.x = Arow[ka];
    a.y = Arow[ka + 1];
#pragma unroll
    for (int nt = 0; nt < 4; ++nt) {
      const int n = nt * 16 + r;
      v2f b;
      if (TRANSB) { b.x = W[(size_t)n * KA + ka];  b.y = W[(size_t)n * KA + ka + 1]; }
      else        { b.x = W[(size_t)ka * DIM + n]; b.y = W[(size_t)(ka + 1) * DIM + n]; }
      acc[nt] = __builtin_amdgcn_wmma_f32_16x16x4_f32(
          false, a, false, b, (short)0, acc[nt], false, false);
    }
  }

#pragma unroll
  for (int nt = 0; nt < 4; ++nt) {
    const int col = nt * 16 + r;
    const float bv = BIAS ? bias[col] : 0.0f;
#pragma unroll
    for (int v = 0; v < 8; ++v) {
      const int row = tile * 16 + kh * 8 + v;
      float val = acc[nt][v] + bv;
      if (LEAKY) val = leaky(val);
      C[(size_t)row * DIM + col] = val;
    }
  }
}

// ---------------------------------------------------------------------------
// Fused layer:  Out = act( [ H | leaky(X @ lin_w^T + lin_b) ] @ g_w^T + g_b )
// H already contains leaky(conv).  One wave per 16-row tile; the 16x128
// concatenated A-matrix is staged through LDS (row stride 132 floats to
// avoid bank conflicts on the transposed fragment reads).
// ---------------------------------------------------------------------------
#define LROW 132
template<bool FINAL_LEAKY>
__global__ __launch_bounds__(128) void layer_wmma(
    const float* __restrict__ X, const float* __restrict__ H,
    const float* __restrict__ lin_w, const float* __restrict__ lin_b,
    const float* __restrict__ g_w,  const float* __restrict__ g_b,
    float* __restrict__ Out, int ntiles)
{
  __shared__ float lds[4][16][LROW];          // 4 waves * 8.25 KB
  const int wave = threadIdx.x >> 5;
  const int lane = threadIdx.x & 31;
  int tile = blockIdx.x * 4 + wave;
  if (tile >= ntiles) tile = ntiles - 1;
  const int r  = lane & 15;
  const int kh = lane >> 4;

  // ---- stage 1a: xhat = leaky(X_tile @ lin_w^T + lin_b) -> LDS cols [64,128)
  {
    const float* Arow = X + (size_t)(tile * 16 + r) * DIM;
    v8f acc[4];
#pragma unroll
    for (int nt = 0; nt < 4; ++nt)
#pragma unroll
      for (int j = 0; j < 8; ++j) acc[nt][j] = 0.0f;
#pragma unroll
    for (int kk = 0; kk < DIM; kk += 4) {
      const int ka = kk + kh * 2;
      v2f a; a.x = Arow[ka]; a.y = Arow[ka + 1];
#pragma unroll
      for (int nt = 0; nt < 4; ++nt) {
        const int n = nt * 16 + r;
        v2f b; b.x = lin_w[(size_t)n * DIM + ka]; b.y = lin_w[(size_t)n * DIM + ka + 1];
        acc[nt] = __builtin_amdgcn_wmma_f32_16x16x4_f32(
            false, a, false, b, (short)0, acc[nt], false, false);
      }
    }
#pragma unroll
    for (int nt = 0; nt < 4; ++nt) {
      const int col = nt * 16 + r;
      const float bv = lin_b[col];
#pragma unroll
      for (int v = 0; v < 8; ++v)
        lds[wave][kh * 8 + v][DIM + col] = leaky(acc[nt][v] + bv);
    }
  }

  // ---- stage 1b: copy H tile (already leaky'd) -> LDS cols [0,64)
  {
    const float* Ht = H + (size_t)tile * 16 * DIM;
#pragma unroll
    for (int i = 0; i < 8; ++i) {
      const int f   = i * 32 + lane;     // 256 float4 slots in a 16x64 tile
      const int row = f >> 4;
      const int c4  = f & 15;
      const float4 v = *(const float4*)(Ht + (size_t)row * DIM + c4 * 4);
      *(float4*)&lds[wave][row][c4 * 4] = v;
    }
  }
  __syncthreads();

  // ---- stage 2: Out = act( A2[16x128] @ g_w^T + g_b )
  v8f acc[4];
#pragma unroll
  for (int nt = 0; nt < 4; ++nt)
#pragma unroll
    for (int j = 0; j < 8; ++j) acc[nt][j] = 0.0f;
#pragma unroll
  for (int kk = 0; kk < 2 * DIM; kk += 4) {
    const int ka = kk + kh * 2;
    v2f a; a.x = lds[wave][r][ka]; a.y = lds[wave][r][ka + 1];
#pragma unroll
    for (int nt = 0; nt < 4; ++nt) {
      const int n = nt * 16 + r;
      v2f b; b.x = g_w[(size_t)n * (2 * DIM) + ka]; b.y = g_w[(size_t)n * (2 * DIM) + ka + 1];
      acc[nt] = __builtin_amdgcn_wmma_f32_16x16x4_f32(
          false, a, false, b, (short)0, acc[nt], false, false);
    }
  }
#pragma unroll
  for (int nt = 0; nt < 4; ++nt) {
    const int col = nt * 16 + r;
    const float bv = g_b[col];
#pragma unroll
    for (int v = 0; v < 8; ++v) {
      const int row = tile * 16 + kh * 8 + v;
      float val = acc[nt][v] + bv;
      if (FINAL_LEAKY) val = leaky(val);
      Out[(size_t)row * DIM + col] = val;
    }
  }
}

// ---------------------------------------------------------------------------
// Helpers
// ---------------------------------------------------------------------------
__global__ void copy_vec4(const float4* __restrict__ src, float4* __restrict__ dst, int n4) {
  int i = blockIdx.x * blockDim.x + threadIdx.x;
  if (i < n4) dst[i] = src[i];
}

__global__ void zero_vec4(float4* __restrict__ dst, int n4) {
  int i = blockIdx.x * blockDim.x + threadIdx.x;
  if (i < n4) dst[i] = make_float4(0.f, 0.f, 0.f, 0.f);
}

// L2-normalize 64-float rows; one wave per row (wave32).
__global__ void normalize_rows(float* __restrict__ X, int nrows) {
  const int row  = blockIdx.x * (blockDim.x >> 5) + (threadIdx.x >> 5);
  const int lane = threadIdx.x & 31;
  if (row >= nrows) return;
  float2 v = *(const float2*)&X[(size_t)row * DIM + lane * 2];
  float ss = v.x * v.x + v.y * v.y;
#pragma unroll
  for (int off = 16; off; off >>= 1) ss += __shfl_xor(ss, off, 32);
  const float scale = 1.0f / fmaxf(sqrtf(ss), 1e-12f);
  v.x *= scale; v.y *= scale;
  *(float2*)&X[(size_t)row * DIM + lane * 2] = v;
}

// S[dst[e]] += X[src[e]] over all edges.  16 lanes per edge, float4 gather,
// native global_atomic_add_f32 (everything is L2-resident: 25.6MB tables).
__global__ void scatter_x(const float* __restrict__ X, const int* __restrict__ src,
                          const int* __restrict__ dst, float* __restrict__ S, int nE) {
  const long t = (long)blockIdx.x * blockDim.x + threadIdx.x;
  const int  e = (int)(t >> 4);
  if (e >= nE) return;
  const int q = (int)(t & 15);
  const int si = src[e];
  const int di = dst[e];
  const float4 v = *(const float4*)&X[(size_t)si * DIM + q * 4];
  float* o = &S[(size_t)di * DIM + q * 4];
  unsafeAtomicAdd(o + 0, v.x);
  unsafeAtomicAdd(o + 1, v.y);
  unsafeAtomicAdd(o + 2, v.z);
  unsafeAtomicAdd(o + 3, v.w);
}

// ---------------------------------------------------------------------------
// Host side
// ---------------------------------------------------------------------------
extern "C" void kernel_launch(void* const* d_in, const int* in_sizes, int n_in,
                              void* d_out, int out_size, void* d_ws, size_t ws_size,
                              hipStream_t stream) {
  // dict order: features, edge_index, preference, mlp_w, mlp_b,
  //             conv_w[0..4], lin_w[0..4], lin_b[0..4], g_w[0..4], g_b[0..4]
  const float* features   = (const float*)d_in[0];
  const int*   edge_index = (const int*)  d_in[1];
  const float* preference = (const float*)d_in[2];
  const float* mlp_w      = (const float*)d_in[3];
  const float* mlp_b      = (const float*)d_in[4];
  const float *conv_w[5], *lin_w[5], *lin_b[5], *g_w[5], *g_b[5];
  for (int i = 0; i < 5; ++i) {
    conv_w[i] = (const float*)d_in[5 + i];
    lin_w[i]  = (const float*)d_in[10 + i];
    lin_b[i]  = (const float*)d_in[15 + i];
    g_w[i]    = (const float*)d_in[20 + i];
    g_b[i]    = (const float*)d_in[25 + i];
  }
  const int* esrc = edge_index;
  const int* edst = edge_index + NEDGE;

  float* x = (float*)d_ws;                 // 3 x 25.6 MB rotating buffers
  float* s = x + (size_t)NTOT * DIM;
  float* h = s + (size_t)NTOT * DIM;
  float* outmu = (float*)d_out;
  float* outlv = outmu + (size_t)NTOT * DIM;

  const int ntiles = NTOT / 16;            // 6250
  const int itiles = NUM_ITEM / 16;        // 3125
  const int n4     = NTOT * DIM / 4;
  const int zgrid  = (n4 + 255) / 256;
  const long sthreads = (long)NEDGE * 16;
  const int sgrid  = (int)((sthreads + 255) / 256);
  const int ggrid  = (ntiles + 7) / 8;
  const int lgrid  = (ntiles + 3) / 4;

  // x = normalize(concat(preference, features @ mlp_w^T + mlp_b))
  {
    const int p4 = NUM_USER * DIM / 4;
    copy_vec4<<<(p4 + 255) / 256, 256, 0, stream>>>((const float4*)preference, (float4*)x, p4);
    gemm_wmma<DIM_FEAT, true, false, true><<<(itiles + 7) / 8, 256, 0, stream>>>(
        features, mlp_w, mlp_b, x + (size_t)NUM_USER * DIM, itiles);
    normalize_rows<<<(NTOT + 7) / 8, 256, 0, stream>>>(x, NTOT);
  }

  // layers 1..3 — note: segsum((x@W)[src]) == segsum(x[src]) @ W, so we
  // scatter raw x and apply conv_w afterwards.
  for (int i = 0; i < 3; ++i) {
    zero_vec4<<<zgrid, 256, 0, stream>>>((float4*)s, n4);
    scatter_x<<<sgrid, 256, 0, stream>>>(x, esrc, edst, s, NEDGE);
    gemm_wmma<DIM, false, true, false><<<ggrid, 256, 0, stream>>>(
        s, conv_w[i], nullptr, h, ntiles);                    // h = leaky(S @ conv_w)
    layer_wmma<true><<<lgrid, 128, 0, stream>>>(
        x, h, lin_w[i], lin_b[i], g_w[i], g_b[i], s, ntiles); // new x -> s
    float* t = x; x = s; s = t;
  }

  // mu / logvar heads share one scatter of the final x (4 scatters total, not 5)
  zero_vec4<<<zgrid, 256, 0, stream>>>((float4*)s, n4);
  scatter_x<<<sgrid, 256, 0, stream>>>(x, esrc, edst, s, NEDGE);

  gemm_wmma<DIM, false, true, false><<<ggrid, 256, 0, stream>>>(
      s, conv_w[3], nullptr, h, ntiles);
  layer_wmma<false><<<lgrid, 128, 0, stream>>>(
      x, h, lin_w[3], lin_b[3], g_w[3], g_b[3], outmu, ntiles);

  gemm_wmma<DIM, false, true, false><<<ggrid, 256, 0, stream>>>(
      s, conv_w[4], nullptr, h, ntiles);
  layer_wmma<false><<<lgrid, 128, 0, stream>>>(
      x, h, lin_w[4], lin_b[4], g_w[4], g_b[4], outlv, ntiles);
}